// PointerNetGenerator_18760417148947
// MI455X (gfx1250) — compile-verified
//
#include <hip/hip_runtime.h>
#include <math.h>

// ---------------------------------------------------------------------------
// Pointer-net generator scoring for MI455X (gfx1250).
// B=8, MEM_LEN=512 (M+1=513), DEC_LEN=32, MEM_H=DEC_H=512, H=1024.
// fp32 throughout; GEMMs via V_WMMA_F32_16X16X4_F32; tanh via v_tanh_f32;
// LDS staging via GLOBAL_LOAD_ASYNC_TO_LDS_B128 (ASYNCcnt tracked).
// ---------------------------------------------------------------------------

typedef float v2f __attribute__((ext_vector_type(2)));
typedef float v8f __attribute__((ext_vector_type(8)));

#define NBATCH   8
#define DECL     32
#define MEMLEN   512
#define MFULL    513
#define KDIM     512
#define NDIM     1024
#define TILES_N  64

__device__ __forceinline__ float fast_tanh(float x) {
#if __has_builtin(__builtin_amdgcn_tanhf)
    return __builtin_amdgcn_tanhf(x);     // gfx1250 v_tanh_f32
#else
    return tanhf(x);
#endif
}

#if __has_builtin(__builtin_amdgcn_global_load_async_to_lds_b128) && \
    __has_builtin(__builtin_amdgcn_s_wait_asynccnt)
#define HAVE_ASYNC_LDS 1
// Parameter types per hipcc diagnostic: (global int4*, shared int4*, imm, imm)
typedef int v4i_vs __attribute__((vector_size(16)));
typedef __attribute__((address_space(1))) v4i_vs* g_v4i_p;
typedef __attribute__((address_space(3))) v4i_vs* l_v4i_p;
// 16-byte global -> LDS async copy (ASYNCcnt tracked, no VGPR staging)
#define ASYNC_CP16(gsrc, ldst)                                              \
    __builtin_amdgcn_global_load_async_to_lds_b128(                         \
        (g_v4i_p)(gsrc), (l_v4i_p)(ldst), 0, 0)
#endif

// --------------------------- weight transpose ------------------------------
// src: [R][C] row-major  ->  dst: [C][R]   (W[h][k] -> Wt[k][h])
__global__ void transpose_kernel(const float* __restrict__ src,
                                 float* __restrict__ dst, int R, int C) {
    int i = blockIdx.x * blockDim.x + threadIdx.x;
    if (i < R * C) {
        int r = i / C, c = i % C;
        dst[(size_t)c * R + r] = src[i];
    }
}

// --------------------------- WMMA fp32 GEMM --------------------------------
// out[b][m][n] = sum_k Arow(b,m)[k] * Wt[k][n] + bias[n]
// Arow(b,0) = term (if term != null), Arow(b,m>0) = A[b][m-1]; else A[b][m].
// One wave computes one 16x16 tile; K-loop of 128 x v_wmma_f32_16x16x4_f32.
// Padded A rows (r >= rows) only feed padded D rows, which are never stored,
// so the row guard is pointer clamping only (no per-iteration scaling).
__global__ void gemm_wmma_kernel(const float* __restrict__ A,
                                 const float* __restrict__ term,
                                 const float* __restrict__ Wt,   // [K][N]
                                 const float* __restrict__ bias, // [N]
                                 float* __restrict__ out,        // [B][rows][N]
                                 int rows, int tilesM, long aBatchStride) {
    const int wave = threadIdx.x >> 5;
    const int lane = threadIdx.x & 31;
    const int half = lane >> 4;      // 0: lanes 0-15, 1: lanes 16-31
    const int l16  = lane & 15;

    int tile = blockIdx.x * 8 + wave;
    int n_t  = tile % TILES_N;
    int m_t  = (tile / TILES_N) % tilesM;
    int b    = tile / (TILES_N * tilesM);

    // ---- A row pointer (fold in terminate_state row; clamp padded rows) ----
    int r  = m_t * 16 + l16;
    int rc = (r < rows) ? r : 0;     // clamp: padded rows read row 0 safely
    const float* arow;
    if (term) {
        arow = (rc == 0) ? term
                         : (A + (size_t)b * aBatchStride + (size_t)(rc - 1) * KDIM);
    } else {
        arow = A + (size_t)b * aBatchStride + (size_t)rc * KDIM;
    }

    const int n0  = n_t * 16;
    const int col = n0 + l16;

    v8f acc = {};
    for (int k = 0; k < KDIM; k += 4) {
        // A 16x4 layout: vgpr j of lane-half h holds K = k + 2*h + j
        int ka = k + half * 2;
        v2f a = *(const v2f*)(arow + ka);
        // B 4x16 layout (mirrors A): vgpr j holds row K = k + 2*half + j
        v2f bb;
        bb.x = Wt[(size_t)ka * NDIM + col];
        bb.y = Wt[(size_t)(ka + 1) * NDIM + col];
#if __has_builtin(__builtin_amdgcn_wmma_f32_16x16x4_f32)
        acc = __builtin_amdgcn_wmma_f32_16x16x4_f32(
            /*neg_a=*/false, a, /*neg_b=*/false, bb,
            /*c_mod=*/(short)0, acc, /*reuse_a=*/false, /*reuse_b=*/false);
#else
        acc[0] += a.x * bb.x + a.y * bb.y;   // compile-survival fallback
#endif
    }

    // ---- store: D layout M = v + 8*half, N = lane%16 ----
    const float bv = bias[col];
#pragma unroll
    for (int v = 0; v < 8; ++v) {
        int mrow = m_t * 16 + v + 8 * half;
        if (mrow < rows)
            out[((size_t)b * rows + mrow) * NDIM + col] = acc[v] + bv;
    }
}

// --------------------- fused tanh + score reduction ------------------------
// raw[b][d][m] = sum_h tanh(pm[b][m][h] + pd[b][d][h]) * w_score[h]
// One wave per m (8 m per block), loop over d. pm rows + w_score in LDS,
// staged with async global->LDS copies.
__global__ void score_kernel(const float* __restrict__ pm,   // [B][513][1024]
                             const float* __restrict__ pd,   // [B][32][1024]
                             const float* __restrict__ wsc,  // [1024]
                             float* __restrict__ raw) {      // [B][32][513]
    __shared__ __align__(16) float ws[NDIM];
    __shared__ __align__(16) float pml[8][NDIM];

    const int tid  = threadIdx.x;
    const int wave = tid >> 5;
    const int lane = tid & 31;
    const int b  = blockIdx.x / 65;
    const int mc = blockIdx.x % 65;
    const int m  = mc * 8 + wave;

    const int msafe = (m < MFULL) ? m : 0;
    const float* pmrow = pm + ((size_t)b * MFULL + msafe) * NDIM;

#ifdef HAVE_ASYNC_LDS
    // w_score: 1024 floats = 256 threads x 16B granules
    ASYNC_CP16(wsc + tid * 4, &ws[tid * 4]);
    // pm row for this wave: 1024 floats = 32 lanes x 8 x 16B granules
#pragma unroll
    for (int i = 0; i < 8; ++i) {
        int off = (i * 32 + lane) * 4;
        ASYNC_CP16(pmrow + off, &pml[wave][off]);
    }
    __builtin_amdgcn_s_wait_asynccnt(0);
    __syncthreads();
#else
    for (int i = tid; i < NDIM; i += 256) ws[i] = wsc[i];
    for (int i = lane; i < NDIM; i += 32) pml[wave][i] = pmrow[i];
    __syncthreads();
#endif

    if (m >= MFULL) return;

    float* rawb = raw + (size_t)b * DECL * MFULL + m;
    for (int d = 0; d < DECL; ++d) {
        const float* pdrow = pd + ((size_t)b * DECL + d) * NDIM;
        if (d + 1 < DECL)   // stage next pd row (global_prefetch_b8)
            __builtin_prefetch(pdrow + NDIM + lane * 32, 0, 0);
        float acc = 0.0f;
#pragma unroll 4
        for (int h = lane; h < NDIM; h += 32)
            acc += fast_tanh(pml[wave][h] + pdrow[h]) * ws[h];
        for (int o = 16; o; o >>= 1) acc += __shfl_xor(acc, o, 32);
        if (lane == 0) rawb[(size_t)d * MFULL] = acc;
    }
}

// --------------------- masking + log_softmax over m ------------------------
__global__ void logsoftmax_kernel(const float* __restrict__ raw,
                                  const int* __restrict__ mem_mask,  // [B][512]
                                  const int* __restrict__ dec_mask,  // [B][32]
                                  const int* __restrict__ dup_mask,  // [B][32][513]
                                  const float* __restrict__ b_score,
                                  float* __restrict__ out) {
    const float NEG_INF = -__builtin_huge_valf();
    const int row  = blockIdx.x;          // b*32 + d, 256 rows
    const int b    = row >> 5;
    const int tid  = threadIdx.x;
    const int lane = tid & 31;
    const int wave = tid >> 5;
    __shared__ float red[8];

    const float bs = b_score[0];
    const int   dm = dec_mask[row];       // decoder-step padding flag

    float v[3];
#pragma unroll
    for (int j = 0; j < 3; ++j) {
        int m = tid + j * 256;
        float s = NEG_INF;
        if (m < MFULL) {
            s = raw[(size_t)row * MFULL + m] + bs;
            if (m > 0 && mem_mask[b * MEMLEN + m - 1]) s = NEG_INF;
            if (!dm && dup_mask[(size_t)row * MFULL + m]) s = NEG_INF;
        }
        v[j] = s;
    }

    // block max
    float mx = fmaxf(fmaxf(v[0], v[1]), v[2]);
    for (int o = 16; o; o >>= 1) mx = fmaxf(mx, __shfl_xor(mx, o, 32));
    if (lane == 0) red[wave] = mx;
    __syncthreads();
    float gmax = red[0];
#pragma unroll
    for (int i = 1; i < 8; ++i) gmax = fmaxf(gmax, red[i]);
    __syncthreads();

    // block sum of exp
    float sum = __expf(v[0] - gmax) + __expf(v[1] - gmax) + __expf(v[2] - gmax);
    for (int o = 16; o; o >>= 1) sum += __shfl_xor(sum, o, 32);
    if (lane == 0) red[wave] = sum;
    __syncthreads();
    float gsum = 0.0f;
#pragma unroll
    for (int i = 0; i < 8; ++i) gsum += red[i];
    const float lse = logf(gsum);

#pragma unroll
    for (int j = 0; j < 3; ++j) {
        int m = tid + j * 256;
        if (m < MFULL)
            out[(size_t)row * MFULL + m] = v[j] - gmax - lse;
    }
}

// ---------------------------------------------------------------------------
extern "C" void kernel_launch(void* const* d_in, const int* in_sizes, int n_in,
                              void* d_out, int out_size, void* d_ws, size_t ws_size,
                              hipStream_t stream) {
    const float* mem      = (const float*)d_in[0];
    const float* dec_hid  = (const float*)d_in[1];
    const int*   mem_mask = (const int*)d_in[2];
    const int*   dec_mask = (const int*)d_in[3];
    const int*   dup_mask = (const int*)d_in[4];
    const float* term     = (const float*)d_in[5];
    const float* W_mem    = (const float*)d_in[6];
    const float* b_mem    = (const float*)d_in[7];
    const float* W_dec    = (const float*)d_in[8];
    const float* b_dec    = (const float*)d_in[9];
    const float* w_score  = (const float*)d_in[10];
    const float* b_score  = (const float*)d_in[11];
    float*       out      = (float*)d_out;

    // workspace layout (floats)
    float* ws      = (float*)d_ws;
    float* Wt_mem  = ws;                                 // 512*1024
    float* Wt_dec  = Wt_mem + (size_t)KDIM * NDIM;       // 512*1024
    float* pm      = Wt_dec + (size_t)KDIM * NDIM;       // 8*513*1024
    float* pd      = pm + (size_t)NBATCH * MFULL * NDIM; // 8*32*1024
    float* raw     = pd + (size_t)NBATCH * DECL * NDIM;  // 8*32*513

    // 1) transpose weights -> [K][N] for coalesced WMMA B loads
    {
        int n = NDIM * KDIM;
        transpose_kernel<<<(n + 255) / 256, 256, 0, stream>>>(W_mem, Wt_mem, NDIM, KDIM);
        transpose_kernel<<<(n + 255) / 256, 256, 0, stream>>>(W_dec, Wt_dec, NDIM, KDIM);
    }

    // 2) pm = [term; mem] @ W_mem^T + b_mem   (8 * 33 * 64 tiles, 8 waves/block)
    gemm_wmma_kernel<<<(NBATCH * 33 * TILES_N) / 8, 256, 0, stream>>>(
        mem, term, Wt_mem, b_mem, pm, MFULL, 33, (long)MEMLEN * KDIM);

    // 3) pd = dec_hid @ W_dec^T + b_dec       (8 * 2 * 64 tiles)
    gemm_wmma_kernel<<<(NBATCH * 2 * TILES_N) / 8, 256, 0, stream>>>(
        dec_hid, nullptr, Wt_dec, b_dec, pd, DECL, 2, (long)DECL * KDIM);

    // 4) fused tanh + score dot (134M v_tanh_f32 + FMA)
    score_kernel<<<NBATCH * 65, 256, 0, stream>>>(pm, pd, w_score, raw);

    // 5) masking + log_softmax over the 513-way axis
    logsoftmax_kernel<<<NBATCH * DECL, 256, 0, stream>>>(
        raw, mem_mask, dec_mask, dup_mask, b_score, out);
}